// MyPostHead_37014028157705
// MI455X (gfx1250) — compile-verified
//
#include <hip/hip_runtime.h>
#include <math.h>

// Problem constants (fixed by the reference)
#define B_   2
#define A_   4
#define C_   3
#define H_   32
#define W_   32
#define HW_  1024            // H*W
#define N_   4096            // H*W*A
#define EPSI 1e-6f

typedef float v2f __attribute__((ext_vector_type(2)));
typedef float v8f __attribute__((ext_vector_type(8)));

// Workspace layout (floats):
//   [0, B*N*4)            : BEV corner boxes  [B][N][4]
//   [B*N*4, +B*C*N)       : exp(scores) in (b,c,n) layout
#define WS_BOXES 0
#define WS_ESC   (B_ * N_ * 4)

// LDS byte offsets (dynamic LDS starts at 0; no static __shared__ elsewhere)
#define LDS_BOX   0                        // N_ float4      = 65536 B
#define LDS_AREA  (LDS_BOX + N_ * 16)      // N_ floats      = 16384 B
#define LDS_V     (LDS_AREA + N_ * 4)      // N_ floats      = 16384 B
#define LDS_P1    (LDS_V + N_ * 4)         // HW_ floats     = 4096 B
#define LDS_RED   (LDS_P1 + HW_ * 4)       // TPB floats
#define LDS_YV    (LDS_RED + 256 * 4)      // WAVES*16 floats
#define LDS_SV    (LDS_YV + 8 * 16 * 4)    // WAVES*16 floats
#define LDS_TOTAL (LDS_SV + 8 * 16 * 4 + 64)

// ---------------------------------------------------------------------------
// Kernel 1: decode BEV boxes + exp(scores) staging
// ---------------------------------------------------------------------------
__global__ void __launch_bounds__(256)
prep_kernel(const float* __restrict__ scores,
            const float* __restrict__ decoded,
            float* __restrict__ ws) {
    int idx = blockIdx.x * blockDim.x + threadIdx.x;   // over B*N
    if (idx >= B_ * N_) return;
    int b = idx / N_;
    int n = idx - b * N_;

    const float* d = decoded + (size_t)(b * N_ + n) * 7;
    float x = d[0], y = d[1], w = d[3], h = d[4], rot = d[6];
    const float PI = 3.14159265358979323846f;
    float normed = fabsf(rot - floorf(rot / PI + 0.5f) * PI);
    bool cond = normed > (0.25f * PI);
    float dw = cond ? h : w;
    float dh = cond ? w : h;

    float* bx = ws + WS_BOXES + (size_t)(b * N_ + n) * 4;
    bx[0] = x - 0.5f * dw;
    bx[1] = y - 0.5f * dh;
    bx[2] = x + 0.5f * dw;
    bx[3] = y + 0.5f * dh;

    // scores (B, A*C, H, W): n = (h*W + w)*A + a
    int a  = n & (A_ - 1);
    int hw = n >> 2;
    float* esc = ws + WS_ESC;
#pragma unroll
    for (int c = 0; c < C_; ++c) {
        float s = scores[((size_t)(b * A_ * C_) + a * C_ + c) * HW_ + hw];
        esc[((size_t)(b * C_) + c) * N_ + n] = __expf(s);
    }
}

// CDNA5 async global->LDS copy of 16 bytes (ASYNCcnt-tracked, no VGPR data).
__device__ __forceinline__ void async_b128(unsigned lds_off, const void* gptr) {
    unsigned long long ga = (unsigned long long)(uintptr_t)gptr;
    asm volatile("global_load_async_to_lds_b128 %0, %1, off"
                 :
                 : "v"(lds_off), "v"(ga)
                 : "memory");
}
__device__ __forceinline__ void wait_async0() {
    asm volatile("s_wait_asynccnt 0x0" ::: "memory");
}

// ---------------------------------------------------------------------------
// Kernel 2: fused IoU + rank-1 bias + softmax-weighted matvec via WMMA
// grid = (rowblocks=32, C, B), block = 256 (8 waves x wave32)
// Each wave owns a 16-row tile; j loops over all 4096 columns in chunks of 4.
// WMMA f32 16x16x4:  A = exp tile (16x4), B col0 = v_j, col1 = 1  =>
// D col0 = numerator, D col1 = denominator, accumulated across the j loop.
// ---------------------------------------------------------------------------
#define WAVES 8
#define TPB   (WAVES * 32)
#define ROWS_PER_WG (WAVES * 16)      // 128
#define RB    (N_ / ROWS_PER_WG)      // 32 row-blocks per (b,c)

__global__ void __launch_bounds__(TPB)
head_kernel(const float* __restrict__ pp,      // pp_params (B, C*3, H, W)
            const float* __restrict__ ws,
            float* __restrict__ out) {
    extern __shared__ float smem[];
    float4* box4_s = (float4*)smem;                       // @ LDS_BOX
    float*  area_s = (float*)((char*)smem + LDS_AREA);
    float*  v_s    = (float*)((char*)smem + LDS_V);
    float*  p1_s   = (float*)((char*)smem + LDS_P1);
    float*  red    = (float*)((char*)smem + LDS_RED);
    float*  yv     = (float*)((char*)smem + LDS_YV);
    float*  sv     = (float*)((char*)smem + LDS_SV);

    const int rb = blockIdx.x;
    const int c  = blockIdx.y;
    const int b  = blockIdx.z;
    const int t  = threadIdx.x;

    const float4* boxes_g = (const float4*)(ws + WS_BOXES) + (size_t)b * N_;
    const float*  esc_g   = ws + WS_ESC + ((size_t)(b * C_) + c) * N_;
    const float*  p0_g    = pp + ((size_t)(b * C_ * 3) + (c * 3 + 0)) * HW_;
    const float*  p1_g    = pp + ((size_t)(b * C_ * 3) + (c * 3 + 1)) * HW_;

    // --- Async-stage the entire j-side into LDS (Tensor/async data path) ---
#pragma unroll
    for (int k = t; k < N_; k += TPB)                 // boxes: 4096 x float4
        async_b128(LDS_BOX + (unsigned)k * 16u, boxes_g + k);
#pragma unroll
    for (int k = t; k < N_ / 4; k += TPB)             // values: 1024 x float4
        async_b128(LDS_V + (unsigned)k * 16u, (const float4*)esc_g + k);
    {
        int k = t;                                    // p1: 256 x float4
        if (k < HW_ / 4)
            async_b128(LDS_P1 + (unsigned)k * 16u, (const float4*)p1_g + k);
    }
    wait_async0();
    __syncthreads();

    // Pass 2 from LDS: box areas + S_sc partial sums
    for (int k = t; k < N_; k += TPB) {
        float4 bb = box4_s[k];
        area_s[k] = (bb.z - bb.x) * (bb.w - bb.y);
    }
    float psum = 0.f;
    for (int k = t; k < N_; k += TPB) psum += v_s[k];
    red[t] = psum;
    __syncthreads();
    for (int s = TPB / 2; s > 0; s >>= 1) {
        if (t < s) red[t] += red[t + s];
        __syncthreads();
    }
    const float Ssc = red[0];

    const int wave = t >> 5;
    const int lane = t & 31;
    const int m    = lane & 15;   // row within tile; also N column index in B/D
    const int half = lane >> 4;   // K half (A: VGPR0 holds K=2*half, VGPR1 K=2*half+1)

    const int ibase = rb * ROWS_PER_WG + wave * 16;
    const int i     = ibase + m;

    // Row-side data, hoisted out of the j loop
    const float4 bi = box4_s[i];
    const float  area_i = area_s[i];
    const float  p0 = p0_g[i >> 2];

    // Branch-free B-column masks: col0 = v_j, col1 = 1.0, others 0
    const float selv = (m == 0) ? 1.f : 0.f;
    const float sel1 = (m == 1) ? 1.f : 0.f;

    v8f acc = {};
#pragma unroll 4
    for (int j0 = 0; j0 < N_; j0 += 4) {
        const int ja = j0 + 2 * half;        // this lane covers K = 2*half, 2*half+1
        const float4 bj0 = box4_s[ja];
        const float4 bj1 = box4_s[ja + 1];
        const float  aj0 = area_s[ja];
        const float  aj1 = area_s[ja + 1];
        const float  p1j = p1_s[j0 >> 2];    // all 4 j's in chunk share hw (A=4)

        // IoU(i, ja)  — fast reciprocal instead of exact divide
        float ltx = fmaxf(bi.x, bj0.x), lty = fmaxf(bi.y, bj0.y);
        float rbx = fminf(bi.z, bj0.z), rby = fminf(bi.w, bj0.w);
        float iw  = fmaxf(rbx - ltx, 0.f), ih = fmaxf(rby - lty, 0.f);
        float inter0 = iw * ih;
        float iou0 = inter0 * __builtin_amdgcn_rcpf(
                         fmaxf(area_i + aj0 - inter0, EPSI));

        // IoU(i, ja+1)
        ltx = fmaxf(bi.x, bj1.x); lty = fmaxf(bi.y, bj1.y);
        rbx = fminf(bi.z, bj1.z); rby = fminf(bi.w, bj1.w);
        iw  = fmaxf(rbx - ltx, 0.f); ih = fmaxf(rby - lty, 0.f);
        float inter1 = iw * ih;
        float iou1 = inter1 * __builtin_amdgcn_rcpf(
                         fmaxf(area_i + aj1 - inter1, EPSI));

        // exp args bounded (iou in [0,1], |p0*p1| small) => no max-sub needed
        const float e0 = __expf(fmaf(p0, p1j, iou0));
        const float e1 = __expf(fmaf(p0, p1j, iou1));

        const float va = v_s[ja], vb = v_s[ja + 1];
        v2f Am = { e0, e1 };                               // A(16x4) exp tile
        v2f Bm = { fmaf(selv, va, sel1),                   // B col0=v, col1=1
                   fmaf(selv, vb, sel1) };

        // D(16x16) += A(16x4) x B(4x16)  — col0: numerator, col1: denominator
        acc = __builtin_amdgcn_wmma_f32_16x16x4_f32(
            /*neg_a=*/false, Am, /*neg_b=*/false, Bm,
            /*c_mod=*/(short)0, acc, /*reuse_a=*/false, /*reuse_b=*/false);
    }

    // Extract D columns 0 (lanes 0/16) and 1 (lanes 1/17): VGPR r holds rows r / r+8
    float* yw = yv + wave * 16;
    float* sw = sv + wave * 16;
    if (m == 0) {
#pragma unroll
        for (int r = 0; r < 8; ++r) yw[r + 8 * half] = acc[r];
    } else if (m == 1) {
#pragma unroll
        for (int r = 0; r < 8; ++r) sw[r + 8 * half] = acc[r];
    }
    __syncthreads();

    if (lane < 16) {
        const int irow = ibase + lane;
        const float o = yw[lane] *
            __builtin_amdgcn_rcpf(fmaxf(sw[lane] * Ssc, 1e-30f));
        const int a  = irow & (A_ - 1);
        const int hw = irow >> 2;
        out[((size_t)(b * A_ * C_) + a * C_ + c) * HW_ + hw] = o;
    }
}

// ---------------------------------------------------------------------------
// Launch
// ---------------------------------------------------------------------------
extern "C" void kernel_launch(void* const* d_in, const int* in_sizes, int n_in,
                              void* d_out, int out_size, void* d_ws, size_t ws_size,
                              hipStream_t stream) {
    const float* scores = (const float*)d_in[0];   // (B, A*C, H, W)
    const float* bbox   = (const float*)d_in[1];   // (B, A*7, H, W)
    const float* pp     = (const float*)d_in[2];   // (B, C*3, H, W)
    const float* dec    = (const float*)d_in[3];   // (B, N, 7)
    float* out = (float*)d_out;
    float* ws  = (float*)d_ws;

    const size_t n_sc   = (size_t)B_ * A_ * C_ * HW_;   // 24576
    const size_t n_bbox = (size_t)B_ * A_ * 7 * HW_;    // 57344
    const size_t n_pp   = (size_t)B_ * C_ * 3 * HW_;    // 18432

    // Passthrough outputs 2 and 3
    hipMemcpyAsync(out + n_sc,          bbox, n_bbox * sizeof(float),
                   hipMemcpyDeviceToDevice, stream);
    hipMemcpyAsync(out + n_sc + n_bbox, pp,   n_pp * sizeof(float),
                   hipMemcpyDeviceToDevice, stream);

    // Stage: boxes + exp(scores)
    prep_kernel<<<(B_ * N_ + 255) / 256, 256, 0, stream>>>(scores, dec, ws);

    // Main fused kernel
    dim3 grid(RB, C_, B_);
    head_kernel<<<grid, TPB, LDS_TOTAL, stream>>>(pp, ws, out);
}